// ModelNew_68161130987790
// MI455X (gfx1250) — compile-verified
//
#include <hip/hip_runtime.h>

typedef __attribute__((ext_vector_type(2))) float v2f;
typedef __attribute__((ext_vector_type(8))) float v8f;

#define IN_F   4096
#define OUT_F  4096
#define BATCH  16384

// LDS tile row stride (floats): 64 data + 2 pad -> conflict-free strided reads
#define XT_STRIDE 66
#define TILE_FLOATS (16 * XT_STRIDE)

// ---------------------------------------------------------------------------
// Phase 1a: column sums of W.  s[j] = sum_i W[i*IN_F + j]
// 128 blocks x 256 threads; coalesced, deterministic fixed-order reduction.
// ---------------------------------------------------------------------------
__global__ __launch_bounds__(256) void colsum_kernel(const float* __restrict__ W,
                                                     float* __restrict__ s) {
  __shared__ float part[8][32];
  const int c   = threadIdx.x & 31;
  const int rg  = threadIdx.x >> 5;            // 0..7
  const int col = blockIdx.x * 32 + c;

  float acc = 0.0f;
  const float* p = W + (size_t)rg * 512 * IN_F + col;
#pragma unroll 8
  for (int i = 0; i < 512; ++i)
    acc += p[(size_t)i * IN_F];

  part[rg][c] = acc;
  __syncthreads();

  if (rg == 0) {
    float t = part[0][c];
#pragma unroll
    for (int g = 1; g < 8; ++g) t += part[g][c];
    s[col] = t;
  }
}

// ---------------------------------------------------------------------------
// Phase 1b: bias_sum -> slot (single block, deterministic tree reduce)
// ---------------------------------------------------------------------------
__global__ __launch_bounds__(256) void bias_sum_kernel(const float* __restrict__ bias,
                                                       float* __restrict__ slot) {
  __shared__ float red[256];
  float a = 0.0f;
  for (int i = threadIdx.x; i < OUT_F; i += 256) a += bias[i];
  red[threadIdx.x] = a;
  __syncthreads();
  for (int st = 128; st > 0; st >>= 1) {
    if (threadIdx.x < st) red[threadIdx.x] += red[threadIdx.x + st];
    __syncthreads();
  }
  if (threadIdx.x == 0) *slot = red[0];
}

// ---------------------------------------------------------------------------
// Async DMA: global -> LDS, 16 bytes per lane, ASYNCcnt-tracked (CDNA5).
// GV mode: VDST = 32-bit LDS byte offset, VADDR = 64-bit per-lane address.
// ---------------------------------------------------------------------------
__device__ __forceinline__ void async_ld_b128(unsigned lds_off, const float* gaddr) {
  asm volatile("global_load_async_to_lds_b128 %0, %1, off"
               :: "v"(lds_off), "v"(gaddr) : "memory");
}

// ---------------------------------------------------------------------------
// Phase 2: out[r] = x[r,:] . s + bias_sum   via V_WMMA_F32_16X16X4_F32
//
// 8 waves/block, 16 batch rows/wave, double-buffered wave-private LDS tiles
// filled by async-to-LDS DMA; 16 WMMA K=4 steps per 64-wide chunk.
// A = x tile (16x4 layout: lanes 0-15 hold K=0,1; lanes 16-31 hold K=2,3),
// B = weight-sum segment broadcast across all 16 N columns, so every D
// column accumulates the same dot product; lane 0 / lane 16 carry N=0.
// ---------------------------------------------------------------------------
__global__ __launch_bounds__(256) void gemv_wmma_kernel(const float* __restrict__ x,
                                                        const float* __restrict__ sb,
                                                        float* __restrict__ out) {
  __shared__ float s_lds[IN_F];                    // 16 KB, block-shared
  __shared__ float x_lds[8][2][TILE_FLOATS];       // 2 x ~33 KB ping-pong tiles

  const int tid  = threadIdx.x;
  const int wave = tid >> 5;
  const int lane = tid & 31;
  const int m    = lane & 15;     // row-in-tile for A, N for D
  const int half = lane >> 4;     // 0: K=0,1   1: K=2,3

  // cooperative load of the weight-sum vector into LDS
#pragma unroll
  for (int i = 0; i < IN_F / 256; ++i)
    s_lds[i * 256 + tid] = sb[i * 256 + tid];
  __syncthreads();

  const int row_base = blockIdx.x * 128 + wave * 16;
  const int row      = 2 * 0 + half;               // per-pass rows: 2p + half
  (void)row;

  // per-lane LDS byte offsets of this lane's float4 slot in each buffer
  unsigned lds_base[2];
  lds_base[0] = (unsigned)(size_t)&x_lds[wave][0][0];
  lds_base[1] = (unsigned)(size_t)&x_lds[wave][1][0];
  const unsigned lane_slot = (unsigned)((half * XT_STRIDE + m * 4) * 4); // row p=0

  // issue async fill of buffer 0 for chunk 0 (8 x b128 per wave)
#pragma unroll
  for (int p = 0; p < 8; ++p) {
    const int r = 2 * p + half;
    async_ld_b128(lds_base[0] + (unsigned)(r * XT_STRIDE * 4 + m * 16),
                  x + (size_t)(row_base + r) * IN_F + m * 4);
  }

  v8f c = {};

  for (int i = 0; i < IN_F / 64; ++i) {
    const int cur = i & 1;
    const int kc  = i * 64;

    if (i < IN_F / 64 - 1) {
      // prefetch next chunk into the other buffer, then wait for current 8
      const int kn = kc + 64;
#pragma unroll
      for (int p = 0; p < 8; ++p) {
        const int r = 2 * p + half;
        async_ld_b128(lds_base[cur ^ 1] + (unsigned)(r * XT_STRIDE * 4 + m * 16),
                      x + (size_t)(row_base + r) * IN_F + kn + m * 4);
      }
      asm volatile("s_wait_asynccnt 0x8" ::: "memory");
    } else {
      asm volatile("s_wait_asynccnt 0x0" ::: "memory");
    }

    const float* xt = &x_lds[wave][cur][0];

    // ---- 16 x WMMA f32 16x16x4 accumulation over this chunk ----
#pragma unroll
    for (int kk = 0; kk < 64; kk += 4) {
      v2f a, b;
      a.x = xt[m * XT_STRIDE + kk + 2 * half];
      a.y = xt[m * XT_STRIDE + kk + 2 * half + 1];
      b.x = s_lds[kc + kk + 2 * half];             // broadcast across N
      b.y = s_lds[kc + kk + 2 * half + 1];
      c = __builtin_amdgcn_wmma_f32_16x16x4_f32(false, a, false, b,
                                                (short)0, c, false, false);
    }
  }

  // column N=0 of D lives in lane 0 (M=0..7) and lane 16 (M=8..15)
  if (m == 0) {
    const float bsum = sb[IN_F];
#pragma unroll
    for (int r = 0; r < 8; ++r)
      out[row_base + half * 8 + r] = c[r] + bsum;
  }
  (void)lane_slot;
}

// ---------------------------------------------------------------------------
extern "C" void kernel_launch(void* const* d_in, const int* in_sizes, int n_in,
                              void* d_out, int out_size, void* d_ws, size_t ws_size,
                              hipStream_t stream) {
  const float* x    = (const float*)d_in[0];   // [BATCH, IN_F]
  const float* W    = (const float*)d_in[1];   // [OUT_F, IN_F]
  const float* bias = (const float*)d_in[2];   // [OUT_F]
  float* out = (float*)d_out;                  // [BATCH] (x1)
  float* s   = (float*)d_ws;                   // s[0..IN_F-1], bias_sum at s[IN_F]

  colsum_kernel<<<IN_F / 32, 256, 0, stream>>>(W, s);
  bias_sum_kernel<<<1, 256, 0, stream>>>(bias, s + IN_F);
  gemv_wmma_kernel<<<BATCH / 128, 256, 0, stream>>>(x, s, out);
}